// RealEv3_45208825757878
// MI455X (gfx1250) — compile-verified
//
#include <hip/hip_runtime.h>
#include <hip/hip_bf16.h>

typedef __attribute__((ext_vector_type(2))) float v2f;
typedef __attribute__((ext_vector_type(8))) float v8f;

// masks[v][j] for the 11 arity masks (v=0..10), zero-padded to v=11..15, j=6..7.
// v even (<10): prefix mask, 1 iff j <= v/2 ; v odd: suffix mask, 1 iff j > v/2 ; v==10: all ones.
__device__ __forceinline__ float mask_val(int v, int j) {
  if (j > 5 || v > 10) return 0.0f;
  if (v == 10) return 1.0f;
  if ((v & 1) == 0) return (j <= (v >> 1)) ? 1.0f : 0.0f;
  return (j > (v >> 1)) ? 1.0f : 0.0f;
}

// xor-16 across the wave via a single ds_swizzle_b32 (offset[14:10]=xor=16, and=0x1F).
__device__ __forceinline__ float xor16_add(float x) {
  int sw = __builtin_amdgcn_ds_swizzle(__float_as_int(x), 0x401F);
  return x + __int_as_float(sw);
}

__global__ __launch_bounds__(256) void realev3_kernel(
    const int* __restrict__ r_idx,
    const int* __restrict__ e1, const int* __restrict__ e2,
    const int* __restrict__ e3, const int* __restrict__ e4,
    const int* __restrict__ e5, const int* __restrict__ e6,
    const float* __restrict__ E, const float* __restrict__ R,
    const float* __restrict__ R_bias, const float* __restrict__ Rw,
    float* __restrict__ out, int B)
{
  const int lane = threadIdx.x & 31;
  const int wid  = threadIdx.x >> 5;
  const int b    = blockIdx.x * (blockDim.x >> 5) + wid;
  if (b >= B) return;                       // wave-uniform; EXEC stays full for WMMA

  const int k   = lane & 15;                // for C: embedding sub-index; for D': row v
  const int hlf = lane >> 4;                // 0: w=0..7 / k=0..7 ; 1: w=8..15 / k=8..15

  // All gather indices are wave-uniform: pin them to SGPRs so addressing is
  // saddr-based (scalar base + small voffset) instead of per-lane 64-bit math.
  const int r = __builtin_amdgcn_readfirstlane(r_idx[b]);
  int eidx[6];
  eidx[0] = __builtin_amdgcn_readfirstlane(e1[b]);
  eidx[1] = __builtin_amdgcn_readfirstlane(e2[b]);
  eidx[2] = __builtin_amdgcn_readfirstlane(e3[b]);
  eidx[3] = __builtin_amdgcn_readfirstlane(e4[b]);
  eidx[4] = __builtin_amdgcn_readfirstlane(e5[b]);
  eidx[5] = __builtin_amdgcn_readfirstlane(e6[b]);

  // Warm the 6 gathered entity rows (1 KB each): 32 lanes x 32B covers the row.
  #pragma unroll
  for (int j = 0; j < 6; ++j)
    __builtin_prefetch(E + (long)eidx[j] * 256 + lane * 8, 0, 3);

  const float* __restrict__ Rrow = R + (long)r * 1536;

  // C[j][k] = sum_w R[r][j][w][k] * E[e_j][w][k]; full value on every lane after xor-16.
  float Cv[6];
  #pragma unroll
  for (int j = 0; j < 6; ++j) {
    const float* __restrict__ erow = E + (long)eidx[j] * 256;
    const float* __restrict__ rrow = Rrow + j * 256;
    float acc = 0.0f;
    #pragma unroll
    for (int w = 0; w < 8; ++w) {
      const int f = ((w + (hlf << 3)) << 4) + k;   // (w + 8*hlf)*16 + k
      acc = __builtin_fmaf(rrow[f], erow[f], acc);
    }
    Cv[j] = xor16_add(acc);
  }

  // Transposed mask einsum:
  //   D'(16x16) = C^T(16x8 pad) x masks^T(8x16 pad) + biasC
  // A = C^T : lane m=k' holds row k' of C^T == Cv[j] values already lane-local.
  // B = masks^T : B[K=j][N=v] = masks[v][j], computed from the prefix/suffix rule.
  // C-operand = bias broadcast: element (M=k, N=v) = bias[k] -> d[g] = bias[g+8*hlf].
  // Result: lane L, reg g holds inner[v = L&15][k = g + 8*hlf] (bias included).
  const int jb = hlf ? 2 : 0;
  v2f a0, a1, b0, b1;
  a0.x = hlf ? Cv[2] : Cv[0];  a0.y = hlf ? Cv[3] : Cv[1];
  a1.x = hlf ? 0.0f  : Cv[4];  a1.y = hlf ? 0.0f  : Cv[5];   // j=6,7 pad -> 0
  b0.x = mask_val(k, jb);      b0.y = mask_val(k, jb + 1);
  b1.x = mask_val(k, 4 + jb);  b1.y = mask_val(k, 5 + jb);   // j=6,7 pad -> 0

  const float* __restrict__ brow = R_bias + (long)r * 16 + (hlf << 3);
  v8f d;
  #pragma unroll
  for (int g = 0; g < 8; ++g) d[g] = brow[g];

  d = __builtin_amdgcn_wmma_f32_16x16x4_f32(false, a0, false, b0, (short)0, d, false, false);
  d = __builtin_amdgcn_wmma_f32_16x16x4_f32(false, a1, false, b1, (short)0, d, false, false);

  // s[v] = sum_k relu(inner[v][k]): 8 lane-local relus + one xor-16 combine.
  float part = 0.0f;
  #pragma unroll
  for (int g = 0; g < 8; ++g) part += fmaxf(d[g], 0.0f);
  const float sfull = xor16_add(part);     // lane L holds s[L & 15]

  const int si = __float_as_int(sfull);
  const float s0  = __int_as_float(__builtin_amdgcn_readlane(si, 0));
  const float s1  = __int_as_float(__builtin_amdgcn_readlane(si, 1));
  const float s2  = __int_as_float(__builtin_amdgcn_readlane(si, 2));
  const float s3  = __int_as_float(__builtin_amdgcn_readlane(si, 3));
  const float s4  = __int_as_float(__builtin_amdgcn_readlane(si, 4));
  const float s5  = __int_as_float(__builtin_amdgcn_readlane(si, 5));
  const float s6  = __int_as_float(__builtin_amdgcn_readlane(si, 6));
  const float s7  = __int_as_float(__builtin_amdgcn_readlane(si, 7));
  const float s8  = __int_as_float(__builtin_amdgcn_readlane(si, 8));
  const float s9  = __int_as_float(__builtin_amdgcn_readlane(si, 9));
  const float s10 = __int_as_float(__builtin_amdgcn_readlane(si, 10));

  const float* __restrict__ wr = Rw + (long)r * 6;   // uniform address -> s_load
  const float o = wr[0] * s0 * s1
                + wr[1] * s2 * s3
                + wr[2] * s4 * s5
                + wr[3] * s6 * s7
                + wr[4] * s8 * s9
                + wr[5] * s10;

  if (lane == 0) out[b] = o;
}

extern "C" void kernel_launch(void* const* d_in, const int* in_sizes, int n_in,
                              void* d_out, int out_size, void* d_ws, size_t ws_size,
                              hipStream_t stream) {
  const int*   r_idx  = (const int*)d_in[0];
  const int*   e1     = (const int*)d_in[1];
  const int*   e2     = (const int*)d_in[2];
  const int*   e3     = (const int*)d_in[3];
  const int*   e4     = (const int*)d_in[4];
  const int*   e5     = (const int*)d_in[5];
  const int*   e6     = (const int*)d_in[6];
  const float* E      = (const float*)d_in[7];
  const float* R      = (const float*)d_in[8];
  const float* R_bias = (const float*)d_in[9];
  const float* Rw     = (const float*)d_in[10];
  float*       out    = (float*)d_out;

  const int B = in_sizes[0];
  const int wavesPerBlock = 8;                      // 256 threads = 8 wave32
  dim3 block(256);
  dim3 grid((B + wavesPerBlock - 1) / wavesPerBlock);
  realev3_kernel<<<grid, block, 0, stream>>>(r_idx, e1, e2, e3, e4, e5, e6,
                                             E, R, R_bias, Rw, out, B);
}